// SignatureToLogSignature_79946521248234
// MI455X (gfx1250) — compile-verified
//
#include <hip/hip_runtime.h>

#define W       299592   // sum 8^k, k=1..6
#define MW      37448    // levels 1..5
#define NBATCH  64
#define NB_SCAN 293      // ceil(W/1024)

typedef __attribute__((ext_vector_type(2))) float v2f;
typedef __attribute__((ext_vector_type(8))) float v8f;

__constant__ int dOFF[7] = {0, 0, 8, 72, 584, 4680, 37448};
__constant__ int dSZ [7] = {0, 8, 64, 512, 4096, 32768, 262144};

// ---------------------------------------------------------------------------
// Lyndon test: word packed as base-8 digits of `local` (3 bits/letter, first
// letter most significant). Lyndon <=> strictly smaller than every proper
// rotation (handles aperiodicity automatically).
// ---------------------------------------------------------------------------
__device__ __forceinline__ int lyndon_flag(int i) {
    int k, local;
    if      (i < 8)     { k = 1; local = i; }
    else if (i < 72)    { k = 2; local = i - 8; }
    else if (i < 584)   { k = 3; local = i - 72; }
    else if (i < 4680)  { k = 4; local = i - 584; }
    else if (i < 37448) { k = 5; local = i - 4680; }
    else                { k = 6; local = i - 37448; }
    for (int j = 1; j < k; ++j) {
        int sh  = 3 * (k - j);
        int rot = ((local & ((1 << sh) - 1)) << (3 * j)) | (local >> sh);
        if (local >= rot) return 0;
    }
    return 1;
}

// ---------------------------------------------------------------------------
// Scan pass 1: per-block (1024 elems) Lyndon-flag totals
// ---------------------------------------------------------------------------
__global__ __launch_bounds__(256) void k_flags_pass1(int* __restrict__ bsums) {
    __shared__ int sh[256];
    const int tid  = threadIdx.x;
    const int base = blockIdx.x * 1024;
    int s = 0;
    #pragma unroll
    for (int q = 0; q < 4; ++q) {
        int i = base + tid * 4 + q;
        s += (i < W) ? lyndon_flag(i) : 0;
    }
    sh[tid] = s;
    __syncthreads();
    for (int off = 128; off > 0; off >>= 1) {
        if (tid < off) sh[tid] += sh[tid + off];
        __syncthreads();
    }
    if (tid == 0) bsums[blockIdx.x] = sh[0];
}

// Scan pass 2 (tiny): exclusive scan of 293 block sums
__global__ void k_scan_bsums(int* __restrict__ bsums) {
    if (threadIdx.x == 0) {
        int run = 0;
        for (int i = 0; i < NB_SCAN; ++i) { int v = bsums[i]; bsums[i] = run; run += v; }
    }
}

// Scan pass 3: pos[i] = output slot if Lyndon else -1
__global__ __launch_bounds__(256) void k_flags_pass2(const int* __restrict__ bsums,
                                                     int* __restrict__ pos) {
    __shared__ int sh[256];
    const int tid  = threadIdx.x;
    const int base = blockIdx.x * 1024;
    int f[4]; int s = 0;
    #pragma unroll
    for (int q = 0; q < 4; ++q) {
        int i = base + tid * 4 + q;
        f[q] = (i < W) ? lyndon_flag(i) : 0;
        s += f[q];
    }
    sh[tid] = s;
    __syncthreads();
    for (int off = 1; off < 256; off <<= 1) {
        int v = (tid >= off) ? sh[tid - off] : 0;
        __syncthreads();
        sh[tid] += v;
        __syncthreads();
    }
    int excl = sh[tid] - s + bsums[blockIdx.x];
    #pragma unroll
    for (int q = 0; q < 4; ++q) {
        int i = base + tid * 4 + q;
        if (i < W) { pos[i] = f[q] ? excl : -1; excl += f[q]; }
    }
}

// ---------------------------------------------------------------------------
// Per-batch: build M_s = sum_m ((-1)^m/(m+1)) (S^(x)m)_s  (levels 1..5),
// then emit Lyndon entries of L1..L4. One block per batch element.
// ---------------------------------------------------------------------------
__global__ __launch_bounds__(256) void k_small(const float* __restrict__ sig,
                                               const int*   __restrict__ pos,
                                               float* __restrict__ Mg,
                                               float* __restrict__ Tpg,
                                               float* __restrict__ Tng,
                                               float* __restrict__ out, int outw) {
    const int b   = blockIdx.x;
    const int tid = threadIdx.x;
    const float* S  = sig + (size_t)b * W;
    float* Mb   = Mg  + (size_t)b * MW;
    float* bufA = Tpg + (size_t)b * MW;
    float* bufB = Tng + (size_t)b * MW;

    for (int i = tid; i < MW; i += 256) Mb[i] = -0.5f * S[i];   // m=1 term
    __syncthreads();

    const float cm[6] = {0.f, 0.f, 1.f / 3.f, -0.25f, 0.2f, -1.f / 6.f};
    const float* P = S;      // T^(1) = S
    float* dst = bufA;
    for (int m = 2; m <= 5; ++m) {
        for (int s = m; s <= 5; ++s) {
            const int sz = dSZ[s], off = dOFF[s];
            for (int i = tid; i < sz; i += 256) {
                float acc = 0.f;
                for (int j = m - 1; j <= s - 1; ++j) {
                    const int sh   = 3 * (s - j);
                    const int mask = (1 << sh) - 1;
                    acc += P[dOFF[j] + (i >> sh)] * S[dOFF[s - j] + (i & mask)];
                }
                if (m < 5) dst[off + i] = acc;
                Mb[off + i] += cm[m] * acc;
            }
        }
        __syncthreads();
        P   = dst;
        dst = (dst == bufA) ? bufB : bufA;
    }

    // L_k = S_k + sum_s M_s (x) S_{k-s}, k=1..4, with fused Lyndon store
    for (int k = 1; k <= 4; ++k) {
        const int sz = dSZ[k], off = dOFF[k];
        for (int i = tid; i < sz; i += 256) {
            float v = S[off + i];
            for (int s = 1; s <= k - 1; ++s) {
                const int sh   = 3 * (k - s);
                const int mask = (1 << sh) - 1;
                v += Mb[dOFF[s] + (i >> sh)] * S[dOFF[k - s] + (i & mask)];
            }
            const int p = pos[off + i];
            if (p >= 0) out[(size_t)b * outw + p] = v;
        }
    }
}

// ---------------------------------------------------------------------------
// Level-5 output: 32768 entries/batch, 4 FMA each, fused Lyndon store.
// ---------------------------------------------------------------------------
__global__ __launch_bounds__(256) void k_level5(const float* __restrict__ sig,
                                                const int*   __restrict__ pos,
                                                const float* __restrict__ Mg,
                                                float* __restrict__ out, int outw) {
    const int b = blockIdx.y;
    const int i = blockIdx.x * 256 + threadIdx.x;   // < 32768
    const float* S  = sig + (size_t)b * W;
    const float* Mb = Mg  + (size_t)b * MW;
    float v = S[4680 + i];
    #pragma unroll
    for (int s = 1; s <= 4; ++s) {
        const int sh   = 3 * (5 - s);
        const int mask = (1 << sh) - 1;
        v += Mb[dOFF[s] + (i >> sh)] * S[dOFF[5 - s] + (i & mask)];
    }
    const int p = pos[4680 + i];
    if (p >= 0) out[(size_t)b * outw + p] = v;
}

// ---------------------------------------------------------------------------
// Level-6 output via WMMA. View level 6 as 64x4096 (r = l1l2, c = l3..l6),
// one wave per aligned 16x16 tile. s=2,3,4 terms are rank-1 within a tile ->
// packed into one V_WMMA_F32_16X16X4_F32 (K slots 0..2, slot 3 zero).
// s=1, s=5 and the +S6 term are selection-structured -> VALU on the D tile.
// ---------------------------------------------------------------------------
__global__ __launch_bounds__(256) void k_level6(const float* __restrict__ sig,
                                                const int*   __restrict__ pos,
                                                const float* __restrict__ Mg,
                                                float* __restrict__ out, int outw) {
    const int tid  = threadIdx.x;
    const int lane = tid & 31;
    const int t    = blockIdx.x * 8 + (tid >> 5);   // tile id, 65536 total
    const int b    = t >> 10;                       // 1024 tiles per batch
    const int tib  = t & 1023;
    const int r0   = (tib >> 8) << 4;               // 0,16,32,48
    const int c0   = (tib & 255) << 4;              // 0..4080
    const int n    = lane & 15;
    const int h    = lane >> 4;

    const float* S  = sig + (size_t)b * W;
    const float* Mb = Mg  + (size_t)b * MW;
    const float* M1 = Mb;          const float* M2 = Mb + 8;
    const float* M3 = Mb + 72;     const float* M4 = Mb + 584;
    const float* M5 = Mb + 4680;
    const float* S1 = S;           const float* S2 = S + 8;
    const float* S3 = S + 72;      const float* S4 = S + 584;
    const float* S5 = S + 4680;    const float* S6 = S + 37448;

    // A (16x4): A[m,0]=M2[r0+m]  A[m,1]=M3[(r0+m)*8+(c0>>9)]
    //           A[m,2]=M4[(r0+m)*64+(c0>>6)]  A[m,3]=0
    // layout: lanes 0-15 carry K0(K1), lanes 16-31 carry K2(K3); M = lane&15
    const float a_k0 = M2[r0 + n];
    const float a_k1 = M3[(r0 + n) * 8 + (c0 >> 9)];
    const float a_k2 = M4[(r0 + n) * 64 + (c0 >> 6)];
    v2f a;
    a.x = h ? a_k2 : a_k0;
    a.y = h ? 0.0f : a_k1;

    // B (4x16): B[0,c]=S4[c0+c]  B[1,c]=S3[(c0&511)+c]  B[2,c]=S2[(c0&63)+c]  B[3,c]=0
    // layout: VGPR0 = rows {0,2}, VGPR1 = rows {1,3}; N = lane&15
    const float b_k0 = S4[c0 + n];
    const float b_k1 = S3[(c0 & 511) + n];
    const float b_k2 = S2[(c0 & 63) + n];
    v2f bm;
    bm.x = h ? b_k2 : b_k0;
    bm.y = h ? 0.0f : b_k1;

    v8f acc = {};
    acc = __builtin_amdgcn_wmma_f32_16x16x4_f32(
        /*neg_a=*/false, a, /*neg_b=*/false, bm,
        /*c_mod=*/(short)0, acc, /*reuse_a=*/false, /*reuse_b=*/false);

    const int c = c0 + n;
    #pragma unroll
    for (int v = 0; v < 8; ++v) {
        const int r   = r0 + v + 8 * h;             // D layout: row = v + 8*(lane>=16)
        const int idx = (r << 12) | c;              // flat level-6 index
        float val = acc[v] + S6[idx]
                  + M1[r >> 3] * S5[((r & 7) << 12) | c]        // s=1 term
                  + M5[(r << 9) | (c >> 3)] * S1[c & 7];        // s=5 term
        const int p = pos[37448 + idx];
        if (p >= 0) out[(size_t)b * outw + p] = val;
    }
}

// ---------------------------------------------------------------------------
extern "C" void kernel_launch(void* const* d_in, const int* in_sizes, int n_in,
                              void* d_out, int out_size, void* d_ws, size_t ws_size,
                              hipStream_t stream) {
    const float* sig = (const float*)d_in[0];
    float* out = (float*)d_out;
    const int outw = out_size / NBATCH;             // 51360

    auto al = [](size_t x) { return (x + 255) & ~(size_t)255; };
    char* ws = (char*)d_ws;
    size_t o = 0;
    int*   pos   = (int*)(ws + o);   o = al(o + (size_t)W * 4);
    int*   bsums = (int*)(ws + o);   o = al(o + (size_t)NB_SCAN * 4);
    float* Mg    = (float*)(ws + o); o = al(o + (size_t)NBATCH * MW * 4);
    float* Tp    = (float*)(ws + o); o = al(o + (size_t)NBATCH * MW * 4);
    float* Tn    = (float*)(ws + o); o = al(o + (size_t)NBATCH * MW * 4);
    (void)ws_size; (void)in_sizes; (void)n_in;

    hipLaunchKernelGGL(k_flags_pass1, dim3(NB_SCAN), dim3(256), 0, stream, bsums);
    hipLaunchKernelGGL(k_scan_bsums,  dim3(1),       dim3(32),  0, stream, bsums);
    hipLaunchKernelGGL(k_flags_pass2, dim3(NB_SCAN), dim3(256), 0, stream, bsums, pos);
    hipLaunchKernelGGL(k_small,  dim3(NBATCH),      dim3(256), 0, stream, sig, pos, Mg, Tp, Tn, out, outw);
    hipLaunchKernelGGL(k_level5, dim3(128, NBATCH), dim3(256), 0, stream, sig, pos, Mg, out, outw);
    hipLaunchKernelGGL(k_level6, dim3(8192),        dim3(256), 0, stream, sig, pos, Mg, out, outw);
}